// FeaturePyramidAttention_5282809774887
// MI455X (gfx1250) — compile-verified
//
#include <hip/hip_runtime.h>
#include <math.h>

// ---------------- constants ----------------
#define NQ      21760
#define CCH     256
#define HEADS   8
#define HD      32
#define LVLS    4
#define PTS     4
#define NB      2

// GEMM tiling
#define BM      128          // block tile rows
#define BN      64           // block tile cols
#define KC      32           // K chunk staged in LDS
#define XS_STR  36           // padded LDS row stride for X tile (16B-aligned, conflict-free)

// ---- Tensor Data Mover availability / arity detection ----
#if __has_builtin(__builtin_amdgcn_tensor_load_to_lds)
#define HAVE_TDM 1
#if __has_include(<hip/amd_detail/amd_gfx1250_TDM.h>)
#define TDM_6ARG 1          // therock-10.0 headers => 6-arg builtin variant
#endif
#endif

__device__ __constant__ int d_dims[4]   = {128, 64, 32, 16};
__device__ __constant__ int d_starts[4] = {0, 16384, 20480, 21504};

typedef float v2f __attribute__((ext_vector_type(2)));
typedef float v4f __attribute__((ext_vector_type(4)));
typedef float v8f __attribute__((ext_vector_type(8)));
typedef unsigned int u32x4 __attribute__((ext_vector_type(4)));
typedef int i32x4 __attribute__((ext_vector_type(4)));
typedef int i32x8 __attribute__((ext_vector_type(8)));

// ---------------- transpose: (C, hw) -> (hw, C)  (flatten feats into x) ----------------
__global__ void __launch_bounds__(256) transpose_CtoQ(const float* __restrict__ f,
                                                      float* __restrict__ x, int hw) {
  __shared__ float tile[32][33];
  const int bx = blockIdx.x * 32;
  const int by = blockIdx.y * 32;
  const int tx = threadIdx.x;
  const int ty = threadIdx.y;
#pragma unroll
  for (int i = 0; i < 32; i += 8)
    tile[ty + i][tx] = f[(size_t)(by + ty + i) * hw + bx + tx];
  __syncthreads();
#pragma unroll
  for (int i = 0; i < 32; i += 8)
    x[(size_t)(bx + ty + i) * CCH + by + tx] = tile[tx][ty + i];
}

// ---------------- transpose: (hw, C) -> (C, hw)  (unflatten x into outputs) ----------------
__global__ void __launch_bounds__(256) transpose_QtoC(const float* __restrict__ x,
                                                      float* __restrict__ out, int hw) {
  __shared__ float tile[32][33];
  const int bx = blockIdx.x * 32;
  const int by = blockIdx.y * 32;
  const int tx = threadIdx.x;
  const int ty = threadIdx.y;
#pragma unroll
  for (int i = 0; i < 32; i += 8)
    tile[ty + i][tx] = x[(size_t)(bx + ty + i) * CCH + by + tx];
  __syncthreads();
#pragma unroll
  for (int i = 0; i < 32; i += 8)
    out[(size_t)(by + ty + i) * hw + bx + tx] = tile[tx][ty + i];
}

// ---------------- TDM helpers ----------------
#ifdef HAVE_TDM
// Build a 2D-tile D# per CDNA5 ISA (CDNA5 ISA) and issue TENSOR_LOAD_TO_LDS.
// One DMA per executing wave; caller must gate to a single wave and fence with
// s_wait_tensorcnt + workgroup barrier.
__device__ __forceinline__ void tdm_load_tile(const float* gsrc, unsigned lds_off,
                                              int tile_d0, int tile_d1,
                                              unsigned tensor_d0, unsigned tensor_d1,
                                              int row_stride,
                                              int pad_en, int pad_int_code, int pad_amt_code) {
  const unsigned long long ga = (unsigned long long)(size_t)gsrc;
  u32x4 g0;
  g0[0] = 1u;                                                  // count=1, user descriptor
  g0[1] = lds_off;                                             // lds_addr (bytes)
  g0[2] = (unsigned)ga;                                        // global_addr[31:0]
  g0[3] = ((unsigned)(ga >> 32) & 0x01FFFFFFu) | (2u << 30);   // global_addr[56:32] | type=2
  i32x8 g1;
  g1[0] = (2 << 16) |                                          // data_size = 4 bytes
          (pad_en << 20) | (pad_int_code << 22) | (pad_amt_code << 25);
  g1[1] = (int)((tensor_d0 & 0xFFFFu) << 16);                  // tensor_dim0[15:0] @ bits 63:48
  g1[2] = (int)((tensor_d0 >> 16) | ((tensor_d1 & 0xFFFFu) << 16));
  g1[3] = (int)((tensor_d1 >> 16) | ((unsigned)tile_d0 << 16));// tile_dim0 @ bits 127:112
  g1[4] = (int)((unsigned)tile_d1 & 0xFFFFu);                  // tile_dim1; tile_dim2 = 0
  g1[5] = row_stride;                                          // tensor_dim0_stride[31:0]
  g1[6] = 0;                                                   // stride[47:32] | dim1_stride lo
  g1[7] = 0;                                                   // dim1_stride hi
  const i32x4 gz = {0, 0, 0, 0};
#ifdef TDM_6ARG
  const i32x8 gz8 = {0, 0, 0, 0, 0, 0, 0, 0};
  __builtin_amdgcn_tensor_load_to_lds(g0, g1, gz, gz, gz8, 0);
#else
  __builtin_amdgcn_tensor_load_to_lds(g0, g1, gz, gz, 0);
#endif
}
#endif

__device__ __forceinline__ void tdm_wait0() {
#if __has_builtin(__builtin_amdgcn_s_wait_tensorcnt)
  __builtin_amdgcn_s_wait_tensorcnt(0);
#else
  asm volatile("s_wait_tensorcnt 0x0" ::: "memory");
#endif
}

// ---------------- register-staging fallback helpers ----------------
__device__ __forceinline__ void load_x_regs(const float* __restrict__ X, int K,
                                            int m0b, int k0, int tid, v4f r[4]) {
#pragma unroll
  for (int j = 0; j < 4; ++j) {
    const int i = tid + 256 * j;
    const int row = i >> 3;
    const int c4  = i & 7;
    r[j] = *(const v4f*)(X + (size_t)(m0b + row) * K + k0 + c4 * 4);
  }
}
__device__ __forceinline__ void store_x_lds(float* __restrict__ xs, int tid, const v4f r[4]) {
#pragma unroll
  for (int j = 0; j < 4; ++j) {
    const int i = tid + 256 * j;
    const int row = i >> 3;
    const int c4  = i & 7;
    *(v4f*)(xs + row * XS_STR + c4 * 4) = r[j];
  }
}
__device__ __forceinline__ void load_w_regs(const float* __restrict__ W, int N,
                                            int n0, int k0, int tid, v4f r[2]) {
#pragma unroll
  for (int j = 0; j < 2; ++j) {
    const int i = tid + 256 * j;
    const int kr = i >> 4;
    const int c4 = i & 15;
    r[j] = *(const v4f*)(W + (size_t)(k0 + kr) * N + n0 + c4 * 4);
  }
}
__device__ __forceinline__ void store_w_lds(float* __restrict__ wsm, int tid, const v4f r[2]) {
#pragma unroll
  for (int j = 0; j < 2; ++j) {
    const int i = tid + 256 * j;
    const int kr = i >> 4;
    const int c4 = i & 15;
    *(v4f*)(wsm + kr * BN + c4 * 4) = r[j];
  }
}

// 32 back-to-back WMMAs per chunk per wave; fragments from LDS.
// X tile: [BM][XS_STR] padded row-major. W tile: [KC][BN] row-major.
__device__ __forceinline__ void compute_chunk(const float* __restrict__ xs,
                                              const float* __restrict__ wsm,
                                              int wave, int r15, int hi, v8f acc[4]) {
  const float* arow = xs + (wave * 16 + r15) * XS_STR + hi * 2;
#pragma unroll
  for (int kk = 0; kk < KC; kk += 4) {
    const v2f a = *(const v2f*)(arow + kk);                 // A: K pair per half-wave
    const float* bcol = wsm + (kk + 2 * hi) * BN + r15;     // B: {W[k][n], W[k+1][n]}
#pragma unroll
    for (int t = 0; t < 4; ++t) {
      v2f b;
      b.x = bcol[t * 16];
      b.y = bcol[t * 16 + BN];
      acc[t] = __builtin_amdgcn_wmma_f32_16x16x4_f32(
          false, a, false, b, (short)0, acc[t], false, false);
    }
  }
}

// out[M,N] = X[M,K] @ W[K,N] + bias (+ res). grid (M/BM, N/BN), 256 thr = 8 waves.
__global__ void __launch_bounds__(256)
wmma_gemm_kernel(const float* __restrict__ X, const float* __restrict__ W,
                 const float* __restrict__ bias, const float* __restrict__ res,
                 float* __restrict__ out, int N, int K) {
  __shared__ float Xs[2][BM * XS_STR];   // 2 * 18 KB
  __shared__ float Ws[2][KC * BN];       // 2 * 8 KB

  const int tid  = threadIdx.x;
  const int wave = tid >> 5;
  const int lane = tid & 31;
  const int r15  = lane & 15;
  const int hi   = lane >> 4;
  const int m0b  = blockIdx.x * BM;
  const int n0   = blockIdx.y * BN;
  const int nchunks = K / KC;

  v8f acc[4] = {v8f{}, v8f{}, v8f{}, v8f{}};

#ifdef HAVE_TDM
  // ---- TDM-staged pipeline: async tensor DMA double-buffered against WMMA ----
  if (wave == 0) {
    tdm_load_tile(X + (size_t)m0b * K, (unsigned)(size_t)&Xs[0][0],
                  KC, BM, (unsigned)K, 1u << 20, K, /*pad*/1, /*32 dw*/4, /*4 dw*/3);
    tdm_load_tile(W + n0, (unsigned)(size_t)&Ws[0][0],
                  BN, KC, (unsigned)N, 1u << 20, N, 0, 0, 0);
    tdm_wait0();
  }
  __syncthreads();
  for (int c = 0; c < nchunks; ++c) {
    if (c + 1 < nchunks && wave == 0) {     // next chunk's DMA flies during compute
      const int k0 = (c + 1) * KC;
      tdm_load_tile(X + (size_t)m0b * K + k0, (unsigned)(size_t)&Xs[(c + 1) & 1][0],
                    KC, BM, (unsigned)K, 1u << 20, K, 1, 4, 3);
      tdm_load_tile(W + (size_t)k0 * N + n0, (unsigned)(size_t)&Ws[(c + 1) & 1][0],
                    BN, KC, (unsigned)N, 1u << 20, N, 0, 0, 0);
    }
    compute_chunk(Xs[c & 1], Ws[c & 1], wave, r15, hi, acc);
    if (c + 1 < nchunks) {
      if (wave == 0) tdm_wait0();           // DMA done before barrier-signal
      __syncthreads();
    }
  }
#else
  // ---- register-staging fallback (global_load_b128 -> ds_store) ----
  v4f xr[4], wr[2];
  load_x_regs(X, K, m0b, 0, tid, xr);
  load_w_regs(W, N, n0, 0, tid, wr);
  store_x_lds(Xs[0], tid, xr);
  store_w_lds(Ws[0], tid, wr);
  __syncthreads();
  for (int c = 0; c < nchunks; ++c) {
    if (c + 1 < nchunks) {
      load_x_regs(X, K, m0b, (c + 1) * KC, tid, xr);
      load_w_regs(W, N, n0, (c + 1) * KC, tid, wr);
    }
    compute_chunk(Xs[c & 1], Ws[c & 1], wave, r15, hi, acc);
    if (c + 1 < nchunks) {
      store_x_lds(Xs[(c + 1) & 1], tid, xr);
      store_w_lds(Ws[(c + 1) & 1], tid, wr);
      __syncthreads();
    }
  }
#endif

  // C/D layout: VGPR r -> M = m0 + r + 8*hi, N = n0 + t*16 + r15
  const int mrow = m0b + wave * 16 + hi * 8;
#pragma unroll
  for (int t = 0; t < 4; ++t) {
    const int n  = n0 + t * 16 + r15;
    const float bv = bias ? bias[n] : 0.0f;
#pragma unroll
    for (int r = 0; r < 8; ++r) {
      const size_t oi = (size_t)(mrow + r) * N + n;
      float v = acc[t][r] + bv;
      if (res) v += res[oi];
      out[oi] = v;
    }
  }
}

// ---------------- deformable sampling ----------------
// one block per query, one wave per head, lane = head-dim channel (HD==32==wave size)
__global__ void __launch_bounds__(256)
msda_sample_kernel(const float* __restrict__ value, const float* __restrict__ off,
                   const float* __restrict__ attnlin, float* __restrict__ hout) {
  const int head = threadIdx.x >> 5;
  const int lane = threadIdx.x & 31;
  const int q    = blockIdx.x;
  if (q >= NQ) return;

  int ql, qp;
  if (q < 16384)      { ql = 0; qp = q; }
  else if (q < 20480) { ql = 1; qp = q - 16384; }
  else if (q < 21504) { ql = 2; qp = q - 20480; }
  else                { ql = 3; qp = q - 21504; }
  const int qdim   = d_dims[ql];
  const float refx = ((qp % qdim) + 0.5f) / (float)qdim;
  const float refy = ((qp / qdim) + 0.5f) / (float)qdim;

  const float* al = attnlin + (size_t)q * (HEADS * LVLS * PTS) + head * 16;
  float lg[16];
  float mx = -1e30f;
#pragma unroll
  for (int j = 0; j < 16; ++j) { lg[j] = al[j]; mx = fmaxf(mx, lg[j]); }
  float s = 0.0f;
#pragma unroll
  for (int j = 0; j < 16; ++j) { lg[j] = __expf(lg[j] - mx); s += lg[j]; }
  const float inv = 1.0f / s;

  const float* ofp = off + (size_t)q * CCH + head * (LVLS * PTS * 2);
  const int vcol   = head * HD + lane;
  float acc = 0.0f;

#pragma unroll
  for (int l = 0; l < LVLS; ++l) {
    const int d  = d_dims[l];
    const float fd = (float)d;
    const float* vbase = value + (size_t)d_starts[l] * CCH + vcol;
#pragma unroll
    for (int p = 0; p < PTS; ++p) {
      const float ox   = ofp[(l * PTS + p) * 2 + 0];
      const float oy   = ofp[(l * PTS + p) * 2 + 1];
      const float xx   = (refx + ox / fd) * fd - 0.5f;
      const float yy   = (refy + oy / fd) * fd - 0.5f;
      const float x0f  = floorf(xx), y0f = floorf(yy);
      const float lx   = xx - x0f,   ly  = yy - y0f;
      const int   x0   = (int)x0f,   y0  = (int)y0f;
      const float wgt  = lg[l * PTS + p] * inv;

      float sv = 0.0f;
      const float tw[4] = {(1.0f - lx) * (1.0f - ly), lx * (1.0f - ly),
                           (1.0f - lx) * ly,          lx * ly};
      const int  dxs[4] = {0, 1, 0, 1};
      const int  dys[4] = {0, 0, 1, 1};
#pragma unroll
      for (int t = 0; t < 4; ++t) {
        const int xi = x0 + dxs[t], yi = y0 + dys[t];
        if (xi >= 0 && xi < d && yi >= 0 && yi < d)
          sv += tw[t] * vbase[(size_t)(yi * d + xi) * CCH];   // 128B coalesced per wave
      }
      acc += wgt * sv;
    }
  }
  hout[(size_t)q * CCH + vcol] = acc;
}

// ---------------- host-side orchestration ----------------
extern "C" void kernel_launch(void* const* d_in, const int* in_sizes, int n_in,
                              void* d_out, int out_size, void* d_ws, size_t ws_size,
                              hipStream_t stream) {
  const float* feat[4] = {(const float*)d_in[0], (const float*)d_in[1],
                          (const float*)d_in[2], (const float*)d_in[3]};
  const float* W_off  = (const float*)d_in[4];
  const float* b_off  = (const float*)d_in[5];
  const float* W_attn = (const float*)d_in[6];
  const float* b_attn = (const float*)d_in[7];
  const float* W_val  = (const float*)d_in[8];
  const float* b_val  = (const float*)d_in[9];
  const float* W_out  = (const float*)d_in[10];
  const float* b_out  = (const float*)d_in[11];

  const int hw[4]     = {16384, 4096, 1024, 256};
  const int lstart[4] = {0, 16384, 20480, 21504};

  float* ws    = (float*)d_ws;
  float* x0    = ws;                       // NQ*256
  float* x1    = x0    + (size_t)NQ * CCH; // NQ*256
  float* val   = x1    + (size_t)NQ * CCH; // NQ*256
  float* offb  = val   + (size_t)NQ * CCH; // NQ*256
  float* attnb = offb  + (size_t)NQ * CCH; // NQ*128
  float* houtb = attnb + (size_t)NQ * 128; // NQ*256

  for (int l = 0; l < 4; ++l) {
    dim3 grid(hw[l] / 32, CCH / 32), blk(32, 8);
    transpose_CtoQ<<<grid, blk, 0, stream>>>(feat[l], x0 + (size_t)lstart[l] * CCH, hw[l]);
  }

  float* xc = x0;
  float* xn = x1;
  for (int b = 0; b < NB; ++b) {
    const float* Wv = W_val  + (size_t)b * CCH * CCH;
    const float* bv = b_val  + (size_t)b * CCH;
    const float* Wo = W_off  + (size_t)b * CCH * CCH;
    const float* bo = b_off  + (size_t)b * CCH;
    const float* Wa = W_attn + (size_t)b * CCH * 128;
    const float* ba = b_attn + (size_t)b * 128;
    const float* Wu = W_out  + (size_t)b * CCH * CCH;
    const float* bu = b_out  + (size_t)b * CCH;

    wmma_gemm_kernel<<<dim3(NQ / BM, CCH / BN), 256, 0, stream>>>(xc, Wv, bv, nullptr, val,  CCH, CCH);
    wmma_gemm_kernel<<<dim3(NQ / BM, CCH / BN), 256, 0, stream>>>(xc, Wo, bo, nullptr, offb, CCH, CCH);
    wmma_gemm_kernel<<<dim3(NQ / BM, 128 / BN), 256, 0, stream>>>(xc, Wa, ba, nullptr, attnb, 128, CCH);
    msda_sample_kernel<<<NQ, 256, 0, stream>>>(val, offb, attnb, houtb);
    wmma_gemm_kernel<<<dim3(NQ / BM, CCH / BN), 256, 0, stream>>>(houtb, Wu, bu, xc, xn, CCH, CCH);
    float* t = xc; xc = xn; xn = t;
  }

  float* out = (float*)d_out;
  size_t obase = 0;
  for (int l = 0; l < 4; ++l) {
    dim3 grid(hw[l] / 32, CCH / 32), blk(32, 8);
    transpose_QtoC<<<grid, blk, 0, stream>>>(xc + (size_t)lstart[l] * CCH, out + obase, hw[l]);
    obase += (size_t)CCH * hw[l];
  }
}